// PINN2_13675175871186
// MI455X (gfx1250) — compile-verified
//
#include <hip/hip_runtime.h>

typedef __attribute__((ext_vector_type(2))) float v2f;
typedef __attribute__((ext_vector_type(8))) float v8f;

#define HST 65     // LDS row stride for activations (bank-conflict-free A gathers)
#define FST 113    // LDS row stride for f tile in final GEMM

// workspace layout (float offsets) — padded weights, precomputed M
#define WS_WH   0        // [4][64][64]  hidden W2..W5, zero padded
#define WS_BH   16384    // [4][64]      hidden biases
#define WS_WO   16640    // [64][112]    W_out padded
#define WS_BO   23808    // [112]        b_out padded
#define WS_M    23920    // [112][112]   M[k][c] = beta[k] - alpha[c][k], padded
#define WS_W1   36464    // [64]         W1 row padded
#define WS_B1   36528    // [64]         b1 padded

static __device__ __forceinline__ v8f wmma4(v2f a, v2f b, v8f c) {
  // D = A(16x4,f32) * B(4x16,f32) + C(16x16,f32)
  return __builtin_amdgcn_wmma_f32_16x16x4_f32(false, a, false, b, (short)0, c,
                                               false, false);
}

__global__ void pinn_prep(const float* __restrict__ alpha, const float* __restrict__ beta,
                          const float* __restrict__ W1, const float* __restrict__ b1,
                          const float* __restrict__ W2, const float* __restrict__ b2,
                          const float* __restrict__ W3, const float* __restrict__ b3,
                          const float* __restrict__ W4, const float* __restrict__ b4,
                          const float* __restrict__ W5, const float* __restrict__ b5,
                          const float* __restrict__ Wo, const float* __restrict__ bo,
                          float* __restrict__ ws) {
  const int t  = blockIdx.x * blockDim.x + threadIdx.x;
  const int nt = gridDim.x * blockDim.x;
  // hidden weights, 64x64 zero-padded (no dynamic pointer indexing -> no scratch)
  for (int i = t; i < 64 * 64; i += nt) {
    int r = i >> 6, c = i & 63;
    bool in = (r < 50) && (c < 50);
    int s = r * 50 + c;
    ws[WS_WH + 0 * 4096 + i] = in ? W2[s] : 0.f;
    ws[WS_WH + 1 * 4096 + i] = in ? W3[s] : 0.f;
    ws[WS_WH + 2 * 4096 + i] = in ? W4[s] : 0.f;
    ws[WS_WH + 3 * 4096 + i] = in ? W5[s] : 0.f;
  }
  for (int i = t; i < 64; i += nt) {
    bool in = i < 50;
    ws[WS_BH + 0 * 64 + i] = in ? b2[i] : 0.f;
    ws[WS_BH + 1 * 64 + i] = in ? b3[i] : 0.f;
    ws[WS_BH + 2 * 64 + i] = in ? b4[i] : 0.f;
    ws[WS_BH + 3 * 64 + i] = in ? b5[i] : 0.f;
    ws[WS_W1 + i] = in ? W1[i] : 0.f;
    ws[WS_B1 + i] = in ? b1[i] : 0.f;
  }
  for (int i = t; i < 64 * 112; i += nt) {
    int r = i / 112, c = i % 112;
    ws[WS_WO + i] = (r < 50 && c < 100) ? Wo[r * 100 + c] : 0.f;
  }
  for (int i = t; i < 112; i += nt)
    ws[WS_BO + i] = (i < 100) ? bo[i] : 0.f;
  for (int i = t; i < 112 * 112; i += nt) {
    int k = i / 112, c = i % 112;
    ws[WS_M + i] = (k < 100 && c < 100) ? (beta[k] - alpha[c * 100 + k]) : 0.f;
  }
}

__launch_bounds__(64)
__global__ void pinn_main(const float* __restrict__ x, const float* __restrict__ ws,
                          const float* __restrict__ dtp, const float* __restrict__ l1p,
                          const float* __restrict__ l2p, float* __restrict__ out,
                          int N, int ngroups) {
  __shared__ float hbuf[2][4][16][HST];   // 2 waves x 4 Taylor streams x 16 rows
  const int lane  = threadIdx.x & 31;
  const int wid   = threadIdx.x >> 5;
  const int gwave = blockIdx.x * 2 + wid;
  const int nwav  = gridDim.x * 2;
  const int row16 = lane & 15;
  const int hi    = lane >> 4;     // lane group
  const int koff  = hi * 2;        // A/B frag K offset for lanes 16-31
  const int rbase = hi * 8;        // C/D row offset for lanes 16-31

  const float* Wp = ws + WS_WH;
  const float* bp = ws + WS_BH;
  const float* Wo = ws + WS_WO;
  const float* bo = ws + WS_BO;
  const float* Mm = ws + WS_M;
  const float* w1 = ws + WS_W1;
  const float* b1 = ws + WS_B1;

  const float dtv = dtp[0], lam1 = l1p[0], lam2 = l2p[0];
  const v8f z8 = {0.f, 0.f, 0.f, 0.f, 0.f, 0.f, 0.f, 0.f};

  float (*H)[16][HST] = hbuf[wid];

  for (int g = gwave; g < ngroups; g += nwav) {
    const int p0 = g * 16;

    // ---- layer 1 (input dim 1): s = x*W1+b1, s'=W1, s''=s'''=0; then tanh chain
    #pragma unroll
    for (int j = 0; j < 4; ++j) {
      int col = j * 16 + row16;
      float w1c = w1[col], b1c = b1[col];
      #pragma unroll
      for (int v = 0; v < 8; ++v) {
        int row = rbase + v;
        int p = p0 + row; p = (p < N) ? p : (N - 1);
        float xv = x[p];
        float s0 = xv * w1c + b1c;
        float s1 = w1c;
        float t = tanhf(s0);
        float a = 1.f - t * t;
        float d1 = a * s1;
        float d2 = -2.f * t * d1 * s1;
        float d3 = -2.f * d1 * d1 * s1 - 2.f * t * d2 * s1;
        H[0][row][col] = t;  H[1][row][col] = d1;
        H[2][row][col] = d2; H[3][row][col] = d3;
      }
    }

    // ---- hidden layers 2..5: 4 streams x 4 N-tiles, K=64 in 16 WMMA steps
    #pragma unroll
    for (int l = 0; l < 4; ++l) {
      const float* W = Wp + l * 4096;
      const float* b = bp + l * 64;
      v8f acc[4][4];
      #pragma unroll
      for (int s = 0; s < 4; ++s)
        #pragma unroll
        for (int j = 0; j < 4; ++j) acc[s][j] = z8;

      for (int k = 0; k < 16; ++k) {
        v2f af[4];
        #pragma unroll
        for (int s = 0; s < 4; ++s) {
          af[s].x = H[s][row16][4 * k + koff];
          af[s].y = H[s][row16][4 * k + koff + 1];
        }
        int brow = 4 * k + koff;
        #pragma unroll
        for (int j = 0; j < 4; ++j) {
          int col = j * 16 + row16;
          v2f bf;
          bf.x = W[brow * 64 + col];
          bf.y = W[(brow + 1) * 64 + col];
          #pragma unroll
          for (int s = 0; s < 4; ++s) acc[s][j] = wmma4(af[s], bf, acc[s][j]);
        }
      }
      // tanh Taylor chain + writeback
      #pragma unroll
      for (int j = 0; j < 4; ++j) {
        int col = j * 16 + row16;
        float bias = b[col];
        #pragma unroll
        for (int v = 0; v < 8; ++v) {
          int row = rbase + v;
          float s0 = acc[0][j][v] + bias;
          float s1 = acc[1][j][v];
          float s2 = acc[2][j][v];
          float s3 = acc[3][j][v];
          float t = tanhf(s0);
          float a = 1.f - t * t;
          float d1 = a * s1;
          float d2 = a * s2 - 2.f * t * d1 * s1;
          float d3 = a * s3 - 4.f * t * d1 * s2 - 2.f * d1 * d1 * s1 - 2.f * t * d2 * s1;
          H[0][row][col] = t;  H[1][row][col] = d1;
          H[2][row][col] = d2; H[3][row][col] = d3;
        }
      }
    }

    // ---- output layer: u, u_x, u_xxx (streams 0,1,3), q padded to 112 (7 tiles)
    float ukeep[7][8];
    float fkeep[7][8];
    #pragma unroll
    for (int j = 0; j < 7; ++j) {
      v8f cu = z8, cx = z8, c3 = z8;
      for (int k = 0; k < 16; ++k) {
        v2f a0, a1, a3;
        a0.x = H[0][row16][4 * k + koff]; a0.y = H[0][row16][4 * k + koff + 1];
        a1.x = H[1][row16][4 * k + koff]; a1.y = H[1][row16][4 * k + koff + 1];
        a3.x = H[3][row16][4 * k + koff]; a3.y = H[3][row16][4 * k + koff + 1];
        int brow = 4 * k + koff;
        int col = j * 16 + row16;
        v2f bf;
        bf.x = Wo[brow * 112 + col];
        bf.y = Wo[(brow + 1) * 112 + col];
        cu = wmma4(a0, bf, cu);
        cx = wmma4(a1, bf, cx);
        c3 = wmma4(a3, bf, c3);
      }
      int col = j * 16 + row16;
      float bob = bo[col];
      #pragma unroll
      for (int v = 0; v < 8; ++v) {
        float u = cu[v] + bob;
        ukeep[j][v] = u;
        fkeep[j][v] = -lam1 * u * cx[v] - lam2 * c3[v];  // 0 in padded cols
      }
    }

    // stash f into (now dead) LDS activation region for A-fragment gathers
    float* F = &H[0][0][0];   // 16 x FST fits in 4*16*HST
    #pragma unroll
    for (int j = 0; j < 7; ++j) {
      int col = j * 16 + row16;
      #pragma unroll
      for (int v = 0; v < 8; ++v) F[(rbase + v) * FST + col] = fkeep[j][v];
    }

    // ---- final: out = u + dt * (f @ M), K = 112 in 28 WMMA steps
    #pragma unroll
    for (int j = 0; j < 7; ++j) {
      v8f c = z8;
      for (int k = 0; k < 28; ++k) {
        v2f af;
        af.x = F[row16 * FST + 4 * k + koff];
        af.y = F[row16 * FST + 4 * k + koff + 1];
        int brow = 4 * k + koff;
        int col = j * 16 + row16;
        v2f bf;
        bf.x = Mm[brow * 112 + col];
        bf.y = Mm[(brow + 1) * 112 + col];
        c = wmma4(af, bf, c);
      }
      int col = j * 16 + row16;
      if (col < 100) {
        #pragma unroll
        for (int v = 0; v < 8; ++v) {
          int p = p0 + rbase + v;
          if (p < N) out[p * 100 + col] = ukeep[j][v] + dtv * c[v];
        }
      }
    }
  }
}

extern "C" void kernel_launch(void* const* d_in, const int* in_sizes, int n_in,
                              void* d_out, int out_size, void* d_ws, size_t ws_size,
                              hipStream_t stream) {
  (void)n_in; (void)out_size; (void)ws_size;
  const float* x     = (const float*)d_in[0];
  const float* dt    = (const float*)d_in[1];
  const float* alpha = (const float*)d_in[2];
  const float* beta  = (const float*)d_in[3];
  const float* W1    = (const float*)d_in[4];
  const float* b1    = (const float*)d_in[5];
  const float* W2    = (const float*)d_in[6];
  const float* b2    = (const float*)d_in[7];
  const float* W3    = (const float*)d_in[8];
  const float* b3    = (const float*)d_in[9];
  const float* W4    = (const float*)d_in[10];
  const float* b4    = (const float*)d_in[11];
  const float* W5    = (const float*)d_in[12];
  const float* b5    = (const float*)d_in[13];
  const float* Wo    = (const float*)d_in[14];
  const float* bo    = (const float*)d_in[15];
  const float* l1    = (const float*)d_in[16];
  const float* l2    = (const float*)d_in[17];
  float* out = (float*)d_out;
  float* ws  = (float*)d_ws;

  const int N = in_sizes[0];
  const int ngroups = (N + 15) / 16;

  pinn_prep<<<64, 256, 0, stream>>>(alpha, beta, W1, b1, W2, b2, W3, b3,
                                    W4, b4, W5, b5, Wo, bo, ws);
  pinn_main<<<1024, 64, 0, stream>>>(x, ws, dt, l1, l2, out, N, ngroups);
}